// ResidualVectorQuantizer_54485955117654
// MI455X (gfx1250) — compile-verified
//
#include <hip/hip_runtime.h>

// ---------------------------------------------------------------------------
// RVQ eval forward for MI455X (gfx1250, wave32).
// Compute-bound on residual@C.T GEMMs -> routed through V_WMMA_F32_16X16X4_F32
// (native f32 WMMA keeps the argmin bit-faithful to the f32 reference GEMM).
// Code norms are hoisted to a prep kernel so the hot loop is pure loads+WMMA
// (f32 WMMA does not co-execute with VALU on CDNA5, unlike <=16-bit WMMA).
// ---------------------------------------------------------------------------

typedef float v2f __attribute__((ext_vector_type(2)));
typedef float v8f __attribute__((ext_vector_type(8)));

#define D_CODE   256
#define KCODES   1024
#define LEVELS   4
#define NVEC     65536                  // 16 * 4096
#define ROWS     32                     // rows per wave (two 16x16 D tiles)
#define LSTRIDE  260                    // padded LDS row stride (floats) -> 64-bank conflict-free
#define BETA_F   0.25f

#define ZQ_ELEMS   (NVEC * D_CODE)          // 16,777,216
#define IDX_ELEMS  (NVEC * LEVELS)          // 262,144
#define OUT_COMMIT (ZQ_ELEMS + IDX_ELEMS)   // 17,039,360
#define OUT_CB     (OUT_COMMIT + 1)

// ws layout: [0] = squared-residual accumulator, [1 .. 1+L*K) = code norms
__global__ __launch_bounds__(256) void rvq_prep(const float* __restrict__ codebooks,
                                                float* __restrict__ ws) {
  int code = blockIdx.x * blockDim.x + threadIdx.x;   // 0 .. L*K-1
  if (code == 0) ws[0] = 0.0f;
  const float* c = codebooks + (long)code * D_CODE;
  float s = 0.0f;
#pragma unroll 4
  for (int k = 0; k < D_CODE; k += 4) {
    float4 v = *(const float4*)(c + k);
    s += v.x * v.x + v.y * v.y + v.z * v.z + v.w * v.w;
  }
  ws[1 + code] = s;
}

__global__ __launch_bounds__(32) void rvq_main(const float* __restrict__ z,
                                               const float* __restrict__ codebooks,
                                               float* __restrict__ out,
                                               float* __restrict__ ws) {
  __shared__ float rtile[ROWS * LSTRIDE];   // residual tile, padded rows
  __shared__ int   idxsh[ROWS];

  const int  lane  = threadIdx.x;           // 0..31 (wave32)
  const int  nlane = lane & 15;             // column within 16-code chunk
  const int  hgrp  = lane >> 4;             // half-wave group (k-split for A/B frags)
  const long base  = (long)blockIdx.x * ROWS;
  const float* __restrict__ cnorm = ws + 1;

  // ---- load 32 rows of z into LDS as the initial residual (coalesced b128) ----
  for (int it = 0; it < 64; ++it) {
    int idx = it * 32 + lane;               // float4 index
    int row = idx >> 6;                     // 64 float4 per row
    int c4  = (idx & 63) * 4;
    float4 v = *(const float4*)(z + (base + row) * D_CODE + c4);
    *(float4*)(rtile + row * LSTRIDE + c4) = v;
  }
  __syncthreads();

  float sq_sum = 0.0f;                      // sum over levels of ||new residual||^2

  for (int level = 0; level < LEVELS; ++level) {
    const float* cb_level = codebooks + (long)level * KCODES * D_CODE;
    const float* cn_level = cnorm + level * KCODES;

    float best0[8], best1[8];
    int   bidx0[8], bidx1[8];
#pragma unroll
    for (int v = 0; v < 8; ++v) {
      best0[v] = 3.0e38f; best1[v] = 3.0e38f; bidx0[v] = 0; bidx1[v] = 0;
    }

    // ---- sweep 1024 codes in chunks of 16 columns ----
    for (int nb = 0; nb < KCODES; nb += 16) {
      const float* cbp = cb_level + (long)(nb + nlane) * D_CODE + 2 * hgrp;
      const float* ap0 = rtile + nlane * LSTRIDE + 2 * hgrp;
      const float* ap1 = rtile + (16 + nlane) * LSTRIDE + 2 * hgrp;

      // issue the per-column norm load early; latency hidden by the k-loop
      float cc = cn_level[nb + nlane];
      // pull next chunk's B rows toward the caches while this chunk computes
      if (nb + 16 < KCODES) __builtin_prefetch(cbp + 16 * D_CODE, 0, 1);

      v8f acc0 = {};                        // D tile rows 0..15
      v8f acc1 = {};                        // D tile rows 16..31
#pragma unroll 8
      for (int k0 = 0; k0 < D_CODE; k0 += 4) {
        v2f b  = *(const v2f*)(cbp + k0);   // B frag: col n = nb+nlane, k = k0+2h..+1
        v2f a0 = *(const v2f*)(ap0 + k0);   // A frag: row m = nlane,     same k slice
        v2f a1 = *(const v2f*)(ap1 + k0);   // A frag: row m = 16+nlane
        acc0 = __builtin_amdgcn_wmma_f32_16x16x4_f32(false, a0, false, b,
                                                     (short)0, acc0, false, false);
        acc1 = __builtin_amdgcn_wmma_f32_16x16x4_f32(false, a1, false, b,
                                                     (short)0, acc1, false, false);
      }

      int n = nb + nlane;
      // score = ||c||^2 - 2 r.c  (||r||^2 is row-constant: argmin-invariant)
#pragma unroll
      for (int v = 0; v < 8; ++v) {
        float s0 = cc - 2.0f * acc0[v];
        float s1 = cc - 2.0f * acc1[v];
        if (s0 < best0[v]) { best0[v] = s0; bidx0[v] = n; }
        if (s1 < best1[v]) { best1[v] = s1; bidx1[v] = n; }
      }
    }

    // ---- argmin reduction across the 16 lanes of each half-group ----
#pragma unroll
    for (int v = 0; v < 8; ++v) {
#pragma unroll
      for (int off = 1; off < 16; off <<= 1) {
        float s0 = __shfl_xor(best0[v], off, 32);
        int   i0 = __shfl_xor(bidx0[v], off, 32);
        if (s0 < best0[v] || (s0 == best0[v] && i0 < bidx0[v])) { best0[v] = s0; bidx0[v] = i0; }
        float s1 = __shfl_xor(best1[v], off, 32);
        int   i1 = __shfl_xor(bidx1[v], off, 32);
        if (s1 < best1[v] || (s1 == best1[v] && i1 < bidx1[v])) { best1[v] = s1; bidx1[v] = i1; }
      }
    }
    // D-layout: row m = hgrp*8 + v (tile0) / 16 + hgrp*8 + v (tile1)
    if (nlane == 0) {
#pragma unroll
      for (int v = 0; v < 8; ++v) {
        idxsh[hgrp * 8 + v]      = bidx0[v];
        idxsh[16 + hgrp * 8 + v] = bidx1[v];
      }
    }
    __syncthreads();

    // ---- write indices (as float; values <= 1023 exact) ----
    out[ZQ_ELEMS + (base + lane) * LEVELS + level] = (float)idxsh[lane];

    // ---- residual update: r -= C[idx]; accumulate ||r_new||^2 ----
    for (int it = 0; it < 64; ++it) {
      int idx = it * 32 + lane;
      int row = idx >> 6;
      int c4  = (idx & 63) * 4;
      int code = idxsh[row];
      float4 q = *(const float4*)(cb_level + (long)code * D_CODE + c4);
      float4 r = *(float4*)(rtile + row * LSTRIDE + c4);
      r.x -= q.x; r.y -= q.y; r.z -= q.z; r.w -= q.w;
      *(float4*)(rtile + row * LSTRIDE + c4) = r;
      sq_sum += r.x * r.x + r.y * r.y + r.z * r.z + r.w * r.w;
    }
    __syncthreads();
  }

  // ---- z_q = z - final residual (q_ste sum telescopes) ----
  for (int it = 0; it < 64; ++it) {
    int idx = it * 32 + lane;
    int row = idx >> 6;
    int c4  = (idx & 63) * 4;
    float4 zv = *(const float4*)(z + (base + row) * D_CODE + c4);
    float4 r  = *(float4*)(rtile + row * LSTRIDE + c4);
    float4 o;
    o.x = zv.x - r.x; o.y = zv.y - r.y; o.z = zv.z - r.z; o.w = zv.w - r.w;
    *(float4*)(out + (base + row) * D_CODE + c4) = o;
  }

  // ---- wave-reduce squared-residual sum, one atomic per wave ----
  for (int off = 1; off < 32; off <<= 1) sq_sum += __shfl_xor(sq_sum, off, 32);
  if (lane == 0) atomicAdd(ws, sq_sum);
}

__global__ void rvq_final(const float* __restrict__ ws, float* __restrict__ out) {
  if (threadIdx.x == 0 && blockIdx.x == 0) {
    float S    = ws[0];
    float mean = S / (float)(NVEC * D_CODE);  // sum of per-level means
    out[OUT_COMMIT] = BETA_F * mean;          // commitment = beta * sum of means
    out[OUT_CB]     = mean;                   // codebook loss
  }
}

extern "C" void kernel_launch(void* const* d_in, const int* in_sizes, int n_in,
                              void* d_out, int out_size, void* d_ws, size_t ws_size,
                              hipStream_t stream) {
  (void)in_sizes; (void)n_in; (void)out_size; (void)ws_size;
  const float* z  = (const float*)d_in[0];       // [16,4096,256] f32
  const float* cb = (const float*)d_in[1];       // [4,1024,256]  f32
  float* out = (float*)d_out;
  float* ws  = (float*)d_ws;

  rvq_prep<<<(LEVELS * KCODES) / 256, 256, 0, stream>>>(cb, ws);
  rvq_main<<<NVEC / ROWS, 32, 0, stream>>>(z, cb, out, ws);
  rvq_final<<<1, 32, 0, stream>>>(ws, out);
}